// CombinedLoss_59313498358340
// MI455X (gfx1250) — compile-verified
//
#include <hip/hip_runtime.h>

typedef __attribute__((ext_vector_type(2))) float v2f;
typedef __attribute__((ext_vector_type(8))) float v8f;

// One wave32 computes, for a 16-row tile of A, the min over ALL points of B of
// the squared Euclidean distance:  rowmin[m] = max(0, min_n ||A[m]-B[n]||^2).
//
// Cross term -2*a.b comes from V_WMMA_F32_16X16X4_F32 (K=3 padded to 4).
// A-matrix 16x4 f32 VGPR layout (ISA 7.12.2): lanes 0-15: v0=K0,v1=K1;
// lanes 16-31: v0=K2,v1=K3; M = lane&15.  B (4x16) mirrored with N = lane&15.
// D layout: reg r, lane l -> (m = r + 8*(l>=16), n = l&15).
__global__ void nn_rowmin_kernel(const float* __restrict__ A, int P,
                                 const float* __restrict__ B, int Q,
                                 float* __restrict__ rowmin) {
  const int lane = threadIdx.x & 31;
  const int wave = blockIdx.x * (blockDim.x >> 5) + (threadIdx.x >> 5);
  const int m_base = wave * 16;
  if (m_base >= P) return;

  const int i  = lane & 15;
  const int hi = lane >> 4;

  // A-tile point for this lane (lanes 0-15 and 16-31 both map to points 0..15)
  const float ax = A[(m_base + i) * 3 + 0];
  const float ay = A[(m_base + i) * 3 + 1];
  const float az = A[(m_base + i) * 3 + 2];
  const float na_i = ax * ax + ay * ay + az * az;

  v2f amat;
  amat.x = hi ? az : ax;    // K0 (lo) / K2 (hi)
  amat.y = hi ? 0.0f : ay;  // K1 (lo) / K3=pad (hi)

  float t[8];
#pragma unroll
  for (int r = 0; r < 8; ++r) t[r] = 3.0e38f;

  const int ntiles = Q >> 4;
  for (int nt = 0; nt < ntiles; ++nt) {
    const int n_pt = nt * 16 + i;
    const float bx = B[n_pt * 3 + 0];
    const float by = B[n_pt * 3 + 1];
    const float bz = B[n_pt * 3 + 2];
    const float nb = bx * bx + by * by + bz * bz;  // ||b_n||^2, n = lane&15

    v2f bmat;
    bmat.x = hi ? bz : bx;
    bmat.y = hi ? 0.0f : by;

    v8f c = {0.f, 0.f, 0.f, 0.f, 0.f, 0.f, 0.f, 0.f};
    v8f d = __builtin_amdgcn_wmma_f32_16x16x4_f32(
        /*neg_a=*/false, amat, /*neg_b=*/false, bmat,
        /*c_mod=*/(short)0, c, /*reuse_a=*/false, /*reuse_b=*/false);

#pragma unroll
    for (int r = 0; r < 8; ++r) {
      // candidate = ||b||^2 - 2*dot ; ||a||^2 added after the n-loop
      t[r] = fminf(t[r], __builtin_fmaf(-2.0f, d[r], nb));
    }
  }

  // min across the 16 lanes of each half (the n dimension of the tile)
#pragma unroll
  for (int r = 0; r < 8; ++r) {
    float v = t[r];
    v = fminf(v, __shfl_xor(v, 1, 32));
    v = fminf(v, __shfl_xor(v, 2, 32));
    v = fminf(v, __shfl_xor(v, 4, 32));
    v = fminf(v, __shfl_xor(v, 8, 32));
    t[r] = v;
  }

  // add ||a_m||^2 for m = r + 8*hi (gathered from the lane that loaded it);
  // do shuffles before any divergence so all source lanes are active
  float res[8];
#pragma unroll
  for (int r = 0; r < 8; ++r) {
    const float na_m = __shfl(na_i, r + 8 * hi, 32);
    res[r] = fmaxf(na_m + t[r], 0.0f);
  }
  if (i == 0) {  // lane 0 writes m=0..7, lane 16 writes m=8..15
#pragma unroll
    for (int r = 0; r < 8; ++r) rowmin[m_base + 8 * hi + r] = res[r];
  }
}

// Single block: deterministic fixed-order reduction of the four NN-distance
// buffers (mean of sqrt) + MSE term, combined per the reference weights.
__global__ void finalize_kernel(const float* __restrict__ cd_a,  // Pa
                                const float* __restrict__ cd_b,  // Q
                                const float* __restrict__ sd_a,  // Ps
                                const float* __restrict__ sd_b,  // Q
                                const float* __restrict__ pc13,
                                const float* __restrict__ pc2,
                                float* __restrict__ out,
                                int Pa, int Q, int Ps, int Nm) {
  __shared__ float red[256];
  const int t = threadIdx.x;
  float s_cda = 0.f, s_cdb = 0.f, s_sda = 0.f, s_sdb = 0.f, s_mse = 0.f;
  for (int idx = t; idx < Pa; idx += 256) s_cda += sqrtf(cd_a[idx]);
  for (int idx = t; idx < Q; idx += 256) s_cdb += sqrtf(cd_b[idx]);
  for (int idx = t; idx < Ps; idx += 256) s_sda += sqrtf(sd_a[idx]);
  for (int idx = t; idx < Q; idx += 256) s_sdb += sqrtf(sd_b[idx]);
  for (int idx = t; idx < Nm; idx += 256) {
    const float df = pc13[idx] - pc2[idx];
    s_mse = __builtin_fmaf(df, df, s_mse);
  }
  // total = mse + 0.5*(mean(cd dist1)+mean(cd dist2)) + (mean(sd1)+mean(sd2))
  red[t] = 0.5f * (s_cda / (float)Pa + s_cdb / (float)Q)
         + (s_sda / (float)Ps + s_sdb / (float)Q)
         + s_mse / (float)Nm;
  __syncthreads();
  for (int off = 128; off > 0; off >>= 1) {
    if (t < off) red[t] += red[t + off];
    __syncthreads();
  }
  if (t == 0) out[0] = red[0];
}

extern "C" void kernel_launch(void* const* d_in, const int* in_sizes, int n_in,
                              void* d_out, int out_size, void* d_ws, size_t ws_size,
                              hipStream_t stream) {
  const float* pc1_0 = (const float*)d_in[0];  // [4,4096,3]
  const float* pc1_1 = (const float*)d_in[1];  // [4,1024,3]
  const float* pc1_3 = (const float*)d_in[2];  // [4,4096,3]
  const float* pc2   = (const float*)d_in[3];  // [4,4096,3]

  const int Pa = in_sizes[0] / 3;  // 16384
  const int Ps = in_sizes[1] / 3;  // 4096
  const int Q  = in_sizes[3] / 3;  // 16384
  const int Nm = in_sizes[2];      // 49152 (MSE element count)

  float* ws   = (float*)d_ws;
  float* cd_a = ws;            // Pa : NN dist^2 of pc1_0 points into pc2
  float* cd_b = cd_a + Pa;     // Q  : NN dist^2 of pc2 points into pc1_0
  float* sd_a = cd_b + Q;      // Ps : NN dist^2 of pc1_1 points into pc2
  float* sd_b = sd_a + Ps;     // Q  : NN dist^2 of pc2 points into pc1_1

  const int WPB = 8;  // 8 waves (256 threads) per block, one 16-row tile per wave
  auto blocks = [](int P) { return ((P >> 4) + WPB - 1) / WPB; };

  nn_rowmin_kernel<<<blocks(Pa), 256, 0, stream>>>(pc1_0, Pa, pc2, Q, cd_a);
  nn_rowmin_kernel<<<blocks(Q), 256, 0, stream>>>(pc2, Q, pc1_0, Pa, cd_b);
  nn_rowmin_kernel<<<blocks(Ps), 256, 0, stream>>>(pc1_1, Ps, pc2, Q, sd_a);
  nn_rowmin_kernel<<<blocks(Q), 256, 0, stream>>>(pc2, Q, pc1_1, Ps, sd_b);

  finalize_kernel<<<1, 256, 0, stream>>>(cd_a, cd_b, sd_a, sd_b, pc1_3, pc2,
                                         (float*)d_out, Pa, Q, Ps, Nm);
}